// feature_extractor_24352464569113
// MI455X (gfx1250) — compile-verified
//
#include <hip/hip_runtime.h>

// Problem constants (from reference): B,C,H,W = 4,3,544,960 ; filter 5x5
#define FS   5
#define PADR 2
#define NB   4
#define NC   3
#define NH   544
#define NW   960
#define KK   (FS * FS)   // 25

static constexpr long long PLANE   = (long long)NH * NW;               // 522240
static constexpr long long OUT_PER = (long long)NB * KK * NC * PLANE;  // 156672000
static constexpr int HALO = 4;                 // left/right halo, keeps 16B alignment
static constexpr int LROW = NW + 2 * HALO;     // 968 floats per staged row
static constexpr int CH4  = NW / 4;            // 240 float4 chunks per row

typedef float v4f __attribute__((ext_vector_type(4)));
typedef int   v4i __attribute__((vector_size(16)));

typedef __attribute__((address_space(1))) v4i* as1_v4i_ptr;
typedef __attribute__((address_space(3))) v4i* as3_v4i_ptr;

#if defined(__has_builtin)
#  if __has_builtin(__builtin_amdgcn_global_load_async_to_lds_b128)
#    define HAVE_ASYNC_LDS 1
#  endif
#endif

__device__ __forceinline__ void wait_async_zero() {
#if defined(HAVE_ASYNC_LDS)
#  if __has_builtin(__builtin_amdgcn_s_wait_asynccnt)
  __builtin_amdgcn_s_wait_asynccnt(0);
#  else
  asm volatile("s_wait_asynccnt 0" ::: "memory");
#  endif
#endif
}

// One workgroup per (tensor t, b, c, y): async-stage 5 input rows into LDS,
// then write 25 shifted output rows with NT b128 stores.
__global__ __launch_bounds__(256) void shift_extract_kernel(
    const float* __restrict__ left,
    const float* __restrict__ right,
    float* __restrict__ out)
{
  __shared__ float lds[FS * LROW];  // 5 * 968 * 4 = 19,360 B

  int wg = blockIdx.x;
  const int y = wg % NH; wg /= NH;
  const int c = wg % NC; wg /= NC;
  const int b = wg % NB; wg /= NB;
  const int t = wg;  // 0 = left, 1 = right

  const float* __restrict__ src =
      (t ? right : left) + (long long)(b * NC + c) * PLANE;
  float* __restrict__ dst = out + (long long)t * OUT_PER;

  const int tid = (int)threadIdx.x;

  // Zero the halos: indices [0..3] and [964..967] of each of the 5 rows.
  if (tid < FS * 2 * HALO) {
    int r   = tid >> 3;       // tid / 8
    int h   = tid & 7;        // tid % 8
    int off = (h < HALO) ? h : (LROW - 2 * HALO + h);
    lds[r * LROW + off] = 0.0f;
  }

  if (tid < CH4) {
    const int x4 = tid * 4;
    // Stage input rows y-2..y+2 into LDS, 16 bytes per lane per row.
    #pragma unroll
    for (int r = 0; r < FS; ++r) {
      int ys = y + r - PADR;
      float* l = &lds[r * LROW + HALO + x4];  // 16B aligned by construction
      if ((unsigned)ys < (unsigned)NH) {
        const float* g = src + (long long)ys * NW + x4;
#if defined(HAVE_ASYNC_LDS)
        __builtin_amdgcn_global_load_async_to_lds_b128(
            (as1_v4i_ptr)(uintptr_t)g,
            (as3_v4i_ptr)(unsigned)(uintptr_t)l,
            0 /*imm offset*/, 0 /*cpol*/);
#else
        *(v4f*)l = *(const v4f*)g;
#endif
      } else {
        v4f z = {0.f, 0.f, 0.f, 0.f};
        *(v4f*)l = z;  // out-of-bounds row -> zeros
      }
    }
  }

  wait_async_zero();   // drain ASYNCcnt before the barrier
  __syncthreads();

  if (tid < CH4) {
    const int x4 = tid * 4;
    // out[b, k*C+c, y, x] = lds[i][x + j + 2], k = i*5+j
    float* o = dst + ((long long)(b * KK) * NC + c) * PLANE
                   + (long long)y * NW + x4;
    #pragma unroll
    for (int i = 0; i < FS; ++i) {
      // The 5 shifted 16B windows for this chunk all live in lds[i][x4..x4+11]:
      // three aligned b128 LDS loads, then pure register selection per j.
      const float* lr = &lds[i * LROW + x4];
      v4f w0 = *(const v4f*)(lr);
      v4f w1 = *(const v4f*)(lr + 4);
      v4f w2 = *(const v4f*)(lr + 8);
      float w[12] = { w0.x, w0.y, w0.z, w0.w,
                      w1.x, w1.y, w1.z, w1.w,
                      w2.x, w2.y, w2.z, w2.w };
      #pragma unroll
      for (int j = 0; j < FS; ++j) {
        v4f v = { w[2 + j], w[3 + j], w[4 + j], w[5 + j] };
        __builtin_nontemporal_store(v, (v4f*)o);  // write-once: bypass L2 retention
        o += (long long)NC * PLANE;               // next k
      }
    }
  }
}

// one_hot_filter[m,0,0,i,j] = (m == i*5+j)  ->  flat[q] = (q % 26 == 0)
__global__ void one_hot_filter_kernel(float* __restrict__ f) {
  int q = blockIdx.x * blockDim.x + threadIdx.x;
  if (q < KK * KK) f[q] = ((q % (KK + 1)) == 0) ? 1.0f : 0.0f;
}

extern "C" void kernel_launch(void* const* d_in, const int* in_sizes, int n_in,
                              void* d_out, int out_size, void* d_ws, size_t ws_size,
                              hipStream_t stream) {
  const float* left  = (const float*)d_in[0];
  const float* right = (const float*)d_in[1];
  float* out = (float*)d_out;

  // 2 tensors * 4 * 3 * 544 rows = 13,056 workgroups of 256 threads (8 waves)
  dim3 grid(2 * NB * NC * NH);
  shift_extract_kernel<<<grid, dim3(256), 0, stream>>>(left, right, out);

  one_hot_filter_kernel<<<dim3((KK * KK + 255) / 256), dim3(256), 0, stream>>>(
      out + 2 * OUT_PER);
}